// NeuralODE_67929202753591
// MI455X (gfx1250) — compile-verified
//
#include <hip/hip_runtime.h>

typedef float v2f __attribute__((ext_vector_type(2)));
typedef float v8f __attribute__((ext_vector_type(8)));

// CDNA5 hardware tanh (V_TANH_F32, single transcendental op), with fallback.
__device__ __forceinline__ float fast_tanh(float x) {
#if __has_builtin(__builtin_amdgcn_tanhf)
    return __builtin_amdgcn_tanhf(x);
#else
    x = fminf(fmaxf(x, -15.0f), 15.0f);
    float e = __builtin_amdgcn_exp2f(x * -2.8853900817779268f); // e^{-2x}
    return (1.0f - e) * __builtin_amdgcn_rcpf(1.0f + e);
#endif
}

struct f3 { float a, b, c; };
__device__ __forceinline__ f3 operator+(f3 p, f3 q) { return {p.a + q.a, p.b + q.b, p.c + q.c}; }
__device__ __forceinline__ f3 operator*(float s, f3 p) { return {s * p.a, s * p.b, s * p.c}; }

// Per-lane constant context (lives in VGPRs; all loops fully unrolled).
struct Ctx {
    float ax[12];   // A-matrix K0 (lanes<16: W[m][0]) / K2 (lanes>=16: W[m][2] or 0)
    float aA[12];   // bias slope:  lanes>=16: wb[m], lanes<16: 0
    float aB[12];   // bias base:   lanes>=16: bb[m], lanes<16: W[m][1]
    float wo[96];   // output-head weights for this lane's 8 neurons per (head,block)
};

// One RHS evaluation. State mirrored across lane halves (lane l and l+16 hold
// the same sample). B: lanes<16 -> (x0,x1)=K0,K1; lanes>=16 -> (x2,1.0)=K2,K3.
// A per 16-neuron block: lanes<16 -> (W[m][0], W[m][1]); lanes>=16 ->
// (W[m][2]-or-0, wb[m]*t+bb[m]). D: lane<16 -> neurons 0..7 of sample=lane,
// lane>=16 -> neurons 8..15 of sample=lane-16.
__device__ __forceinline__ f3 feval(float t, const f3 s, const Ctx& C, int half) {
    v2f bm;
    bm.x = half ? s.c : s.a;
    bm.y = half ? 1.0f : s.b;
    float acc[3];
#pragma unroll
    for (int h = 0; h < 3; ++h) {
        float a0 = 0.0f, a1 = 0.0f;   // two chains to break fmac dependency
#pragma unroll
        for (int g = 0; g < 4; ++g) {
            const int i = h * 4 + g;
            v2f am;
            am.x = C.ax[i];
            am.y = fmaf(C.aA[i], t, C.aB[i]);   // time-varying bias in K=3 row
            v8f cz = {};
            v8f d = __builtin_amdgcn_wmma_f32_16x16x4_f32(
                false, am, false, bm, (short)0, cz, false, false);
            // All 8 tanhs first: independent back-to-back trans ops (each one
            // covers the previous one's post-trans hazard slot -> no v_nop).
            float th[8];
#pragma unroll
            for (int r = 0; r < 8; ++r) th[r] = fast_tanh(d[r]);
#pragma unroll
            for (int r = 0; r < 8; r += 2) {
                a0 = fmaf(th[r],     C.wo[i * 8 + r],     a0);
                a1 = fmaf(th[r + 1], C.wo[i * 8 + r + 1], a1);
            }
        }
        acc[h] = a0 + a1;
    }
    f3 o;
    o.a = acc[0] + __shfl_xor(acc[0], 16, 32);   // combine neuron halves
    o.b = acc[1] + __shfl_xor(acc[1], 16, 32);
    o.c = acc[2] + __shfl_xor(acc[2], 16, 32);
    return o;
}

__device__ __forceinline__ f3 dopri5(float t, f3 x, float h, const Ctx& C, int half) {
    f3 k1 = feval(t, x, C, half);
    f3 k2 = feval(t + h * 0.2f,        x + h * ((1.0f/5.0f) * k1), C, half);
    f3 k3 = feval(t + h * 0.3f,        x + h * ((3.0f/40.0f) * k1 + (9.0f/40.0f) * k2), C, half);
    f3 k4 = feval(t + h * 0.8f,        x + h * ((44.0f/45.0f) * k1 + (-56.0f/15.0f) * k2 + (32.0f/9.0f) * k3), C, half);
    f3 k5 = feval(t + h * (8.0f/9.0f), x + h * ((19372.0f/6561.0f) * k1 + (-25360.0f/2187.0f) * k2 + (64448.0f/6561.0f) * k3 + (-212.0f/729.0f) * k4), C, half);
    f3 k6 = feval(t + h,               x + h * ((9017.0f/3168.0f) * k1 + (-355.0f/33.0f) * k2 + (46732.0f/5247.0f) * k3 + (49.0f/176.0f) * k4 + (-5103.0f/18656.0f) * k5), C, half);
    return x + h * ((35.0f/384.0f) * k1 + (500.0f/1113.0f) * k3 + (125.0f/192.0f) * k4 + (-2187.0f/6784.0f) * k5 + (11.0f/84.0f) * k6);
}

__global__ void __launch_bounds__(256)
neural_ode_kernel(const float* __restrict__ u,
                  const float* __restrict__ W1, const float* __restrict__ W2,
                  const float* __restrict__ W3,
                  const float* __restrict__ wb1, const float* __restrict__ bb1,
                  const float* __restrict__ wb2, const float* __restrict__ bb2,
                  const float* __restrict__ wb3, const float* __restrict__ bb3,
                  const float* __restrict__ Wo1, const float* __restrict__ Wo2,
                  const float* __restrict__ Wo3,
                  float* __restrict__ out, int B)
{
    const int tid    = blockIdx.x * blockDim.x + threadIdx.x;
    const int lane   = threadIdx.x & 31;
    const int waveId = tid >> 5;
    const int half   = lane >> 4;
    const int mrow   = lane & 15;

    const int s0 = waveId * 16 + mrow;
    const bool ok = s0 < B;
    const int sample = ok ? s0 : 0;

    // ---- Preload per-lane weight context into registers -------------------
    Ctx C;
    const float* Ws[3]  = {W1, W2, W3};
    const float* wbs[3] = {wb1, wb2, wb3};
    const float* bbs[3] = {bb1, bb2, bb3};
    const float* Wos[3] = {Wo1, Wo2, Wo3};
#pragma unroll
    for (int h = 0; h < 3; ++h) {
        const int ic = (h == 2) ? 3 : 2;   // input columns of this head
#pragma unroll
        for (int g = 0; g < 4; ++g) {
            const int m = 16 * g + mrow;   // neuron row for this lane
            const int i = h * 4 + g;
            float w0 = Ws[h][m * ic + 0];
            float w1 = Ws[h][m * ic + 1];
            float w2 = (ic == 3) ? Ws[h][m * ic + 2] : 0.0f;
            C.ax[i] = half ? w2 : w0;
            C.aA[i] = half ? wbs[h][m] : 0.0f;
            C.aB[i] = half ? bbs[h][m] : w1;
#pragma unroll
            for (int r = 0; r < 8; ++r)
                C.wo[i * 8 + r] = Wos[h][16 * g + half * 8 + r];
        }
    }

    // Mirrored state: both lane halves load the same sample.
    f3 x;
    x.a = u[sample * 3 + 0];
    x.b = u[sample * 3 + 1];
    x.c = u[sample * 3 + 2];

    // ---- features: 24 dopri5 steps, h = 10/24 -----------------------------
    f3 xf = x;
    {
        const float hstep = 10.0f / 24.0f;
        float t = 0.0f;
#pragma unroll 1
        for (int i = 0; i < 24; ++i) { xf = dopri5(t, xf, hstep, C, half); t += hstep; }
    }
    if (ok && !half) {
        out[sample * 3 + 0] = xf.a;
        out[sample * 3 + 1] = xf.b;
        out[sample * 3 + 2] = xf.c;
    }

    // ---- trajectory: 9 segments x 3 substeps, h = (10/9)/3 ----------------
    f3 xt = x;
    const float seg = 10.0f / 9.0f;
    const float h2  = seg / 3.0f;
    float* traj = out + (size_t)B * 3;
#pragma unroll 1
    for (int s = 0; s < 9; ++s) {
        float t = (float)s * seg;
#pragma unroll 1
        for (int j = 0; j < 3; ++j) { xt = dopri5(t, xt, h2, C, half); t += h2; }
        if (ok && !half) {
            size_t o = (size_t)s * (size_t)B * 3 + (size_t)sample * 3;
            traj[o + 0] = xt.a;
            traj[o + 1] = xt.b;
            traj[o + 2] = xt.c;
        }
    }
}

extern "C" void kernel_launch(void* const* d_in, const int* in_sizes, int n_in,
                              void* d_out, int out_size, void* d_ws, size_t ws_size,
                              hipStream_t stream) {
    (void)n_in; (void)out_size; (void)d_ws; (void)ws_size;
    const float* u   = (const float*)d_in[0];
    const float* W1  = (const float*)d_in[1];
    const float* W2  = (const float*)d_in[2];
    const float* W3  = (const float*)d_in[3];
    const float* wb1 = (const float*)d_in[4];
    const float* bb1 = (const float*)d_in[5];
    const float* wb2 = (const float*)d_in[6];
    const float* bb2 = (const float*)d_in[7];
    const float* wb3 = (const float*)d_in[8];
    const float* bb3 = (const float*)d_in[9];
    const float* Wo1 = (const float*)d_in[10];
    const float* Wo2 = (const float*)d_in[11];
    const float* Wo3 = (const float*)d_in[12];

    const int B = in_sizes[0] / 3;          // 131072
    const int threads = 256;                 // 8 waves
    const int samplesPerBlock = (threads / 32) * 16;   // 128
    const int grid = (B + samplesPerBlock - 1) / samplesPerBlock;

    neural_ode_kernel<<<grid, threads, 0, stream>>>(
        u, W1, W2, W3, wb1, bb1, wb2, bb2, wb3, bb3, Wo1, Wo2, Wo3,
        (float*)d_out, B);
}